// GaussianRegLoss_26225070310164
// MI455X (gfx1250) — compile-verified
//
#include <hip/hip_runtime.h>
#include <hip/hip_bf16.h>

typedef __attribute__((ext_vector_type(2))) float v2f;
typedef __attribute__((ext_vector_type(8))) float v8f;

// Problem dims (fixed by setup_inputs)
constexpr int   BS   = 4;
constexpr int   NJ   = 22;
constexpr int   HWD  = 64 * 64 * 64;         // 262144 voxels per batch
constexpr int   WAVES = 8;
constexpr int   TPB   = WAVES * 32;          // 256 threads = 8 wave32
constexpr int   TILES_PER_WAVE = 4;          // 4 x 16-voxel WMMA tiles per wave
constexpr int   VOX_PER_BLOCK  = WAVES * TILES_PER_WAVE * 16;  // 512
constexpr int   BLOCKS_PER_BATCH = HWD / VOX_PER_BLOCK;        // 512
constexpr int   NPART = BLOCKS_PER_BATCH * BS;                 // 2048 partials
constexpr float LOG2E     = 1.4426950408889634f;
constexpr float INV_2SIG2 = 0.02f;           // 1/(2*5^2), applied OUTSIDE the exponent (faithful)

// Joint-tile 1 row map: rows 0..2 -> joints 16..18, rows 8..10 -> joints 19..21,
// all other rows are zero padding.  => valid iff r < 3, uniform across the wave.
__device__ __forceinline__ int tile1_joint_of_row(int m) {
    if (m < 3)              return 16 + m;
    if (m >= 8 && m < 11)   return 11 + m;   // 19 + (m-8)
    return 31;                               // padded slot (zeros in LDS tables)
}

// Stage 1: per-block partial sums of (exp(-d2)/50 - sigmoid(hm))^2
__global__ __launch_bounds__(TPB) void gauss_loss_partial(
    const float* __restrict__ kpts_pred,   // [BS,3,NJ]
    const float* __restrict__ heatmaps,    // [BS,NJ,HWD]
    const float* __restrict__ grids,       // [BS,HWD,3]
    float* __restrict__ partials)          // [BS*BLOCKS_PER_BATCH]
{
    const int b    = blockIdx.y;
    const int tid  = threadIdx.x;
    const int lane = tid & 31;
    const int wave = tid >> 5;
    const bool hi  = lane >= 16;      // lanes 16-31 carry K=2,3 of A/B operands
    const int  l16 = lane & 15;

    // Keypoint components + |c|^2, padded to 32 joints with zeros (slot 31 stays 0)
    __shared__ float sCx[32], sCy[32], sCz[32], sC2[32];
    if (tid < 32) {
        const int j = tid;
        float cx = 0.f, cy = 0.f, cz = 0.f;
        if (j < NJ) {
            cx = kpts_pred[(b * 3 + 0) * NJ + j];
            cy = kpts_pred[(b * 3 + 1) * NJ + j];
            cz = kpts_pred[(b * 3 + 2) * NJ + j];
        }
        sCx[j] = cx; sCy[j] = cy; sCz[j] = cz;
        sC2[j] = cx * cx + cy * cy + cz * cz;
    }
    __syncthreads();

    // A operands: 16x4 (M=joint rows, K = x,y,z,0). ISA layout:
    //   lanes 0-15: VGPR0=K0(x), VGPR1=K1(y); lanes 16-31: VGPR0=K2(z), VGPR1=K3(0)
    v2f A0, A1;
    {
        const int j0 = l16;                       // tile 0: rows 0..15 = joints 0..15
        const int j1 = tile1_joint_of_row(l16);   // tile 1: remapped joints 16..21
        A0.x = hi ? sCz[j0] : sCx[j0];
        A0.y = hi ? 0.f     : sCy[j0];
        A1.x = hi ? sCz[j1] : sCx[j1];
        A1.y = hi ? 0.f     : sCy[j1];
    }

    // This lane's joint indices in each D tile (D row M = r + 8*hi)
    const int joff0 = hi ? 8 : 0;      // tile 0: j = r + joff0
    const int j1b   = hi ? 19 : 16;    // tile 1: j = j1b + r, r < 3

    float c2a[8], c2b[3];
    #pragma unroll
    for (int r = 0; r < 8; ++r) c2a[r] = sC2[r + joff0];
    #pragma unroll
    for (int r = 0; r < 3; ++r) c2b[r] = sC2[j1b + r];

    const float* gbase = grids    + (size_t)b * HWD * 3;
    const float* hbase = heatmaps + (size_t)b * NJ * HWD;

    float acc0 = 0.f, acc1 = 0.f;      // split accumulators (shorter fmac chains)

    for (int t = 0; t < TILES_PER_WAVE; ++t) {
        const int vbase = blockIdx.x * VOX_PER_BLOCK + wave * (TILES_PER_WAVE * 16) + t * 16;
        const int v     = vbase + l16;   // this lane's voxel column (N = lane%16)

        if (t + 1 < TILES_PER_WAVE)      // gfx1250 global_prefetch_b8 for next grid tile
            __builtin_prefetch(&gbase[(size_t)(v + 16) * 3], 0, 1);

        // Issue ALL 11 streaming heatmap loads for this tile up front (MLP),
        // non-temporal: 92 MB dead stream should not churn L2.
        const float* hvb = hbase + v;
        float hv0[8], hv1[3];
        #pragma unroll
        for (int r = 0; r < 8; ++r)
            hv0[r] = __builtin_nontemporal_load(hvb + (size_t)(r + joff0) * HWD);
        #pragma unroll
        for (int r = 0; r < 3; ++r)
            hv1[r] = __builtin_nontemporal_load(hvb + (size_t)(j1b + r) * HWD);

        const float gx = gbase[(size_t)v * 3 + 0];
        const float gy = gbase[(size_t)v * 3 + 1];
        const float gz = gbase[(size_t)v * 3 + 2];
        const float g2 = gx * gx + gy * gy + gz * gz;

        // B operand: 4x16 (K = x,y,z,0 ; N = voxels)
        v2f Bm;
        Bm.x = hi ? gz : gx;
        Bm.y = hi ? 0.f : gy;

        v8f zero = {};
        // D[j, v] = sum_c kpts[c,j] * g[c,v]  (K=3 padded to 4)
        v8f D0 = __builtin_amdgcn_wmma_f32_16x16x4_f32(
            false, A0, false, Bm, (short)0, zero, false, false);
        v8f D1 = __builtin_amdgcn_wmma_f32_16x16x4_f32(
            false, A1, false, Bm, (short)0, zero, false, false);

        #pragma unroll
        for (int r = 0; r < 8; ++r) {
            const float d2    = g2 - 2.f * D0[r] + c2a[r];
            const float gauss = __builtin_amdgcn_exp2f(-d2 * LOG2E) * INV_2SIG2;
            const float sg    = __builtin_amdgcn_rcpf(
                                    1.f + __builtin_amdgcn_exp2f(-hv0[r] * LOG2E));
            const float df    = gauss - sg;
            acc0 += df * df;
        }
        #pragma unroll
        for (int r = 0; r < 3; ++r) {
            const float d2    = g2 - 2.f * D1[r] + c2b[r];
            const float gauss = __builtin_amdgcn_exp2f(-d2 * LOG2E) * INV_2SIG2;
            const float sg    = __builtin_amdgcn_rcpf(
                                    1.f + __builtin_amdgcn_exp2f(-hv1[r] * LOG2E));
            const float df    = gauss - sg;
            acc1 += df * df;
        }
    }

    // Deterministic block reduction
    __shared__ float red[TPB];
    red[tid] = acc0 + acc1;
    __syncthreads();
    #pragma unroll
    for (int s = TPB / 2; s > 0; s >>= 1) {
        if (tid < s) red[tid] += red[tid + s];
        __syncthreads();
    }
    if (tid == 0)
        partials[blockIdx.y * gridDim.x + blockIdx.x] = red[0];
}

// Stage 2: reduce the 2048 partials, apply 0.5/N (LOSS_WEIGHT = 1)
__global__ __launch_bounds__(256) void gauss_loss_finalize(
    const float* __restrict__ partials, float* __restrict__ out)
{
    __shared__ float red[256];
    float s = 0.f;
    for (int i = threadIdx.x; i < NPART; i += 256) s += partials[i];
    red[threadIdx.x] = s;
    __syncthreads();
    #pragma unroll
    for (int st = 128; st > 0; st >>= 1) {
        if (threadIdx.x < st) red[threadIdx.x] += red[threadIdx.x + st];
        __syncthreads();
    }
    if (threadIdx.x == 0) {
        const float n = (float)BS * (float)NJ * (float)HWD;
        out[0] = 0.5f * red[0] / n;
    }
}

extern "C" void kernel_launch(void* const* d_in, const int* in_sizes, int n_in,
                              void* d_out, int out_size, void* d_ws, size_t ws_size,
                              hipStream_t stream) {
    // setup_inputs order: kpts_gt (UNUSED by reference), kpts_pred, heatmaps, grids
    const float* kpts_pred = (const float*)d_in[1];
    const float* heatmaps  = (const float*)d_in[2];
    const float* grids     = (const float*)d_in[3];
    float* out      = (float*)d_out;
    float* partials = (float*)d_ws;   // NPART * 4 = 8 KB scratch

    dim3 grid(BLOCKS_PER_BATCH, BS);
    gauss_loss_partial<<<grid, TPB, 0, stream>>>(kpts_pred, heatmaps, grids, partials);
    gauss_loss_finalize<<<1, 256, 0, stream>>>(partials, out);
}